// GAT_24945170055465
// MI455X (gfx1250) — compile-verified
//
#include <hip/hip_runtime.h>
#include <math.h>

typedef __attribute__((ext_vector_type(2))) float v2f;
typedef __attribute__((ext_vector_type(4))) float v4f;
typedef __attribute__((ext_vector_type(8))) float v8f;

#define HID 64
#define NH  8
#define FEAT (NH * HID)   // 512

// ---------------------------------------------------------------------------
// CSR construction: histogram -> block scan -> fill
// ---------------------------------------------------------------------------
__global__ __launch_bounds__(256) void hist_kernel(const int* __restrict__ dst,
                                                   int* __restrict__ deg, int E) {
    int i = blockIdx.x * 256 + threadIdx.x;
    if (i < E) atomicAdd(&deg[dst[i]], 1);
}

__global__ __launch_bounds__(1024) void scan_kernel(const int* __restrict__ deg,
                                                    int* __restrict__ rowptr,
                                                    int* __restrict__ cursor, int N) {
    __shared__ int sm[1024];
    __shared__ int sbase;
    int t = threadIdx.x;
    if (t == 0) { sbase = 0; rowptr[0] = 0; }
    __syncthreads();
    for (int c = 0; c < N; c += 1024) {
        int i = c + t;
        int v = (i < N) ? deg[i] : 0;
        sm[t] = v;
        __syncthreads();
        for (int off = 1; off < 1024; off <<= 1) {
            int tv = (t >= off) ? sm[t - off] : 0;
            __syncthreads();
            sm[t] += tv;
            __syncthreads();
        }
        int incl = sm[t];
        int base = sbase;
        if (i < N) { rowptr[i + 1] = base + incl; cursor[i] = base + incl - v; }
        __syncthreads();
        if (t == 1023) sbase = base + incl;
        __syncthreads();
    }
}

__global__ __launch_bounds__(256) void fill_kernel(const int* __restrict__ src,
                                                   const int* __restrict__ dst,
                                                   int* __restrict__ cursor,
                                                   int* __restrict__ colsrc, int E) {
    int i = blockIdx.x * 256 + threadIdx.x;
    if (i < E) {
        int d = dst[i];
        int pos = atomicAdd(&cursor[d], 1);
        colsrc[pos] = src[i];
    }
}

// ---------------------------------------------------------------------------
// Dense GEMM via V_WMMA_F32_16X16X4_F32:  Out[n,j] = sum_k X[n,k]*W[j,k] (+bias)
// Block = 256 threads (8 waves); each wave computes a 16-row x 64-col strip.
// B (the 64xK tile of W) is staged in LDS with a bank-conflict-free padded
// stride; the wave's entire A strip is preloaded into registers so the fully
// unrolled WMMA loop depends only on low-latency ds_loads.
// ---------------------------------------------------------------------------
template <int K>
__global__ __launch_bounds__(256) void gemm_wmma_kernel(
    const float* __restrict__ X, const float* __restrict__ W,
    const float* __restrict__ bias, float* __restrict__ Out,
    int n, int J) {
    constexpr int LDW = K + 4;            // (K+4)%64==4 -> conflict-free b64 reads
    __shared__ float sW[64 * LDW];

    int tid  = threadIdx.x;
    int wave = tid >> 5;
    int lane = tid & 31;
    int l16  = lane & 15;
    int hi   = lane >> 4;                 // 0: k pair {0,1}, 1: k pair {2,3}
    int rowBase = blockIdx.x * 128 + wave * 16;
    int colBase = blockIdx.y * 64;

    // Cooperative, coalesced staging of the 64 x K tile of W into LDS.
    for (int idx = tid * 4; idx < 64 * K; idx += 256 * 4) {
        int row = idx / K, col = idx % K;
        *(v4f*)&sW[row * LDW + col] =
            *(const v4f*)&W[(size_t)(colBase + row) * K + col];
    }

    // Preload the wave's entire 16 x K A-strip into registers (K/4 x b64 loads).
    int r = rowBase + l16;
    if (r >= n) r = n - 1;                // clamp loads; stores masked below
    const float* xrow = X + (size_t)r * K;
    v2f av[K / 4];
#pragma unroll
    for (int k0 = 0; k0 < K; k0 += 4) av[k0 / 4] = *(const v2f*)(xrow + k0 + hi * 2);

    __syncthreads();

    const float* s0 = &sW[(0 * 16 + l16) * LDW + hi * 2];
    const float* s1 = &sW[(1 * 16 + l16) * LDW + hi * 2];
    const float* s2 = &sW[(2 * 16 + l16) * LDW + hi * 2];
    const float* s3 = &sW[(3 * 16 + l16) * LDW + hi * 2];

    v8f acc0 = {}, acc1 = {}, acc2 = {}, acc3 = {};
#pragma unroll
    for (int k0 = 0; k0 < K; k0 += 4) {
        v2f a  = av[k0 / 4];
        v2f b0 = *(const v2f*)(s0 + k0);
        v2f b1 = *(const v2f*)(s1 + k0);
        v2f b2 = *(const v2f*)(s2 + k0);
        v2f b3 = *(const v2f*)(s3 + k0);
        acc0 = __builtin_amdgcn_wmma_f32_16x16x4_f32(false, a, false, b0, (short)0, acc0, false, false);
        acc1 = __builtin_amdgcn_wmma_f32_16x16x4_f32(false, a, false, b1, (short)0, acc1, false, false);
        acc2 = __builtin_amdgcn_wmma_f32_16x16x4_f32(false, a, false, b2, (short)0, acc2, false, false);
        acc3 = __builtin_amdgcn_wmma_f32_16x16x4_f32(false, a, false, b3, (short)0, acc3, false, false);
    }

    float bv0 = bias ? bias[colBase + 0 * 16 + l16] : 0.f;
    float bv1 = bias ? bias[colBase + 1 * 16 + l16] : 0.f;
    float bv2 = bias ? bias[colBase + 2 * 16 + l16] : 0.f;
    float bv3 = bias ? bias[colBase + 3 * 16 + l16] : 0.f;
#pragma unroll
    for (int v = 0; v < 8; ++v) {
        int row = rowBase + v + hi * 8;
        if (row < n) {
            float* o = Out + (size_t)row * J + colBase;
            o[0 * 16 + l16] = acc0[v] + bv0;
            o[1 * 16 + l16] = acc1[v] + bv1;
            o[2 * 16 + l16] = acc2[v] + bv2;
            o[3 * 16 + l16] = acc3[v] + bv3;
        }
    }
}

// ---------------------------------------------------------------------------
// el/er: one wave per (node, head). el[n,h] = dot(feat[n,h,:], al[h,:])
// ---------------------------------------------------------------------------
__global__ __launch_bounds__(256) void elr_kernel(
    const float* __restrict__ feat, const float* __restrict__ al,
    const float* __restrict__ ar, float* __restrict__ el,
    float* __restrict__ er, int n) {
    int h = threadIdx.x >> 5;             // wave index = head
    int lane = threadIdx.x & 31;
    int node = blockIdx.x;
    if (node >= n) return;
    const float* f = feat + (size_t)node * FEAT + h * HID;
    float f1 = f[lane], f2 = f[lane + 32];
    float dl = f1 * al[h * HID + lane] + f2 * al[h * HID + 32 + lane];
    float dr = f1 * ar[h * HID + lane] + f2 * ar[h * HID + 32 + lane];
#pragma unroll
    for (int m = 16; m >= 1; m >>= 1) {
        dl += __shfl_xor(dl, m, 32);
        dr += __shfl_xor(dr, m, 32);
    }
    if (lane == 0) { el[node * NH + h] = dl; er[node * NH + h] = dr; }
}

// ---------------------------------------------------------------------------
// Per-dst aggregation: edge softmax + weighted gather + head-sum + bias + leaky
// One 64-thread block per destination node; thread t owns feature f = t of all
// 8 heads (acc[i] pairs with the per-edge attention scalar of head i).
// ---------------------------------------------------------------------------
__global__ __launch_bounds__(64) void agg_kernel(
    const float* __restrict__ feat, const float* __restrict__ el,
    const float* __restrict__ er, const float* __restrict__ bmat,
    const int* __restrict__ rowptr, const int* __restrict__ colsrc,
    float* __restrict__ Y, int n) {
    int node = blockIdx.x;
    int t = threadIdx.x;
    int start = rowptr[node], end = rowptr[node + 1];

    __shared__ float sred[64];
    __shared__ float smax[NH];

    // pass 1: per-head running max over incoming edges (8 threads per head)
    int h = t & 7;
    float erh = er[(size_t)node * NH + h];
    float mx = -3.402823466e38f;
    for (int p = start + (t >> 3); p < end; p += 8) {
        int s = colsrc[p];
        float e = el[(size_t)s * NH + h] + erh;
        e = e > 0.f ? e : 0.2f * e;
        mx = fmaxf(mx, e);
    }
    sred[t] = mx;
    __syncthreads();
    if (t < 8) {
        float m = sred[t];
        for (int q = 1; q < 8; ++q) m = fmaxf(m, sred[t + 8 * q]);
        smax[t] = m;
    }
    __syncthreads();

    v4f erd0 = *(const v4f*)(er + (size_t)node * NH);
    v4f erd1 = *(const v4f*)(er + (size_t)node * NH + 4);
    float mreg[NH], acc[NH], ssum[NH];
#pragma unroll
    for (int i = 0; i < NH; ++i) { mreg[i] = smax[i]; acc[i] = 0.f; ssum[i] = 0.f; }

    // pass 2: exp-weighted gather (ssum recomputed identically by all threads)
    for (int p = start; p < end; ++p) {
        int s = colsrc[p];
        const float* fs = feat + (size_t)s * FEAT;
        v4f el0 = *(const v4f*)(el + (size_t)s * NH);
        v4f el1 = *(const v4f*)(el + (size_t)s * NH + 4);
#pragma unroll
        for (int i = 0; i < NH; ++i) {
            float e = (i < 4 ? el0[i & 3] : el1[i & 3]) +
                      (i < 4 ? erd0[i & 3] : erd1[i & 3]);
            e = e > 0.f ? e : 0.2f * e;
            float a = __expf(e - mreg[i]);
            ssum[i] += a;
            acc[i]  += a * fs[i * HID + t];
        }
    }

    float y = 0.f;
#pragma unroll
    for (int i = 0; i < NH; ++i) {
        float v = ssum[i] > 0.f ? acc[i] / ssum[i] : 0.f;   // empty-row guard
        y += v + bmat[i * HID + t];
    }
    y = y > 0.f ? y : 0.01f * y;
    Y[(size_t)node * HID + t] = y;
}

// ---------------------------------------------------------------------------
extern "C" void kernel_launch(void* const* d_in, const int* in_sizes, int n_in,
                              void* d_out, int out_size, void* d_ws, size_t ws_size,
                              hipStream_t stream) {
    const float* x0  = (const float*)d_in[0];
    const int*   src = (const int*)d_in[1];
    const int*   dst = (const int*)d_in[2];
    const float* W[3]  = {(const float*)d_in[3],  (const float*)d_in[7],  (const float*)d_in[11]};
    const float* al[3] = {(const float*)d_in[4],  (const float*)d_in[8],  (const float*)d_in[12]};
    const float* ar[3] = {(const float*)d_in[5],  (const float*)d_in[9],  (const float*)d_in[13]};
    const float* bb[3] = {(const float*)d_in[6],  (const float*)d_in[10], (const float*)d_in[14]};
    const float* Wm = (const float*)d_in[15];
    const float* bm = (const float*)d_in[16];

    int N = in_sizes[0] / 128;
    int E = in_sizes[1];

    char* p = (char*)d_ws;
    auto walloc = [&](size_t bytes) {
        char* q = p;
        p += (bytes + 255) & ~(size_t)255;
        return q;
    };
    int*   deg    = (int*)  walloc((size_t)N * 4);
    int*   rowptr = (int*)  walloc((size_t)(N + 1) * 4);
    int*   cursor = (int*)  walloc((size_t)N * 4);
    int*   colsrc = (int*)  walloc((size_t)E * 4);
    float* feat   = (float*)walloc((size_t)N * FEAT * 4);
    float* el     = (float*)walloc((size_t)N * NH * 4);
    float* er     = (float*)walloc((size_t)N * NH * 4);
    float* ybufA  = (float*)walloc((size_t)N * HID * 4);
    float* ybufB  = (float*)walloc((size_t)N * HID * 4);

    // CSR by destination (rebuilt every call: deterministic work, no leftover state)
    hipMemsetAsync(deg, 0, (size_t)N * 4, stream);
    hist_kernel<<<(E + 255) / 256, 256, 0, stream>>>(dst, deg, E);
    scan_kernel<<<1, 1024, 0, stream>>>(deg, rowptr, cursor, N);
    fill_kernel<<<(E + 255) / 256, 256, 0, stream>>>(src, dst, cursor, colsrc, E);

    float* ybufs[2] = {ybufA, ybufB};
    const float* xin = x0;
    for (int l = 0; l < 3; ++l) {
        dim3 g((N + 127) / 128, FEAT / 64);
        if (l == 0)
            gemm_wmma_kernel<128><<<g, 256, 0, stream>>>(xin, W[l], nullptr, feat, N, FEAT);
        else
            gemm_wmma_kernel<64><<<g, 256, 0, stream>>>(xin, W[l], nullptr, feat, N, FEAT);
        elr_kernel<<<N, 256, 0, stream>>>(feat, al[l], ar[l], el, er, N);
        float* Y = ybufs[l & 1];
        agg_kernel<<<N, 64, 0, stream>>>(feat, el, er, bb[l], rowptr, colsrc, Y, N);
        xin = Y;
    }

    dim3 gf((N + 127) / 128, 1);
    gemm_wmma_kernel<64><<<gf, 256, 0, stream>>>(xin, Wm, bm, (float*)d_out, N, HID);
}